// ConduitHydrology_13864154431755
// MI455X (gfx1250) — compile-verified
//
#include <hip/hip_runtime.h>
#include <stdint.h>

// ---------------- problem geometry ----------------
#define GRID_ROWS 1024
#define GRID_COLS 1024
#define NNODES (GRID_ROWS * GRID_COLS)
#define N_ITERS 32
#define SEC_PER_A 31556926.0f
#define FLOW_COEFF 0.0405f

// step-kernel tiling: 64 x 16 tile, 66 x 18 halo tile in LDS
#define TX 64
#define TY 16
#define HX (TX + 2)
#define HY (TY + 2)

typedef __attribute__((ext_vector_type(4))) unsigned int u32x4;
typedef __attribute__((ext_vector_type(8))) int          i32x8;
typedef __attribute__((ext_vector_type(4))) int          i32x4;

#if defined(__HIP_DEVICE_COMPILE__)
#if __has_builtin(__builtin_amdgcn_tensor_load_to_lds) && \
    __has_builtin(__builtin_amdgcn_s_wait_tensorcnt)
#define USE_TDM 1
#endif
#endif
#ifndef USE_TDM
#define USE_TDM 0
#endif

// ---------------- hot kernel first (so the disasm snippet shows it) ----------
// One Jacobi flow-accumulation step:
//   q_out = runoff + w.x*q[left] + w.y*q[right] + w.z*q[up] + w.w*q[down]
// q tile (66x18 incl. halo) staged into LDS by the Tensor Data Mover.
__global__ __launch_bounds__(256) void ch_step(
    const float* __restrict__ qin, float* __restrict__ qout,
    const float* __restrict__ runoff, const float4* __restrict__ w4)
{
    __shared__ float tile[HY * HX];   // 66*18*4 = 4752 B
    const int bx = blockIdx.x, by = blockIdx.y;
    const int t  = threadIdx.x;
    const int startCol = bx * TX - 1;   // halo start (may be -1: lands in our
    const int startRow = by * TY - 1;   // own written ws, multiplied by w==0)

#if USE_TDM
    if (t < 32) {   // one wave drives the Tensor Data Mover
        unsigned long long gaddr = (unsigned long long)(uintptr_t)qin
            + (long long)4 * ((long long)startRow * GRID_COLS + (long long)startCol);
        unsigned ldsOff = (unsigned)(uintptr_t)(&tile[0]);  // low 32b = LDS byte offset
        // tensor_dim = remaining extent from tile start -> right/bottom OOB reads 0
        unsigned td0 = (unsigned)(GRID_COLS - startCol);    // <= 1025
        unsigned td1 = (unsigned)(GRID_ROWS - startRow);    // <= 1025
        // D# group 0: count=1 | lds_addr | global_addr[56:0] | type=2
        u32x4 g0 = { 1u,
                     ldsOff,
                     (unsigned)(gaddr & 0xFFFFFFFFull),
                     ((unsigned)((gaddr >> 32) & 0x01FFFFFFull)) | 0x80000000u };
        // D# group 1: data_size=4B | tensor_dim0/1 | tile 66x18 | dim0 stride 1024
        i32x8 g1 = { (int)0x00020000u,                                     // data_size=2 (4B)
                     (int)((td0 & 0xFFFFu) << 16),                         // td0 low
                     (int)(((td0 >> 16) & 0xFFFFu) | ((td1 & 0xFFFFu) << 16)),
                     (int)(((td1 >> 16) & 0xFFFFu) | ((unsigned)HX << 16)),// tile_dim0=66
                     (int)HY,                                              // tile_dim1=18
                     (int)GRID_COLS,                                       // dim0 stride
                     0, 0 };
        i32x4 gz4 = { 0, 0, 0, 0 };
        i32x8 gz8 = { 0, 0, 0, 0, 0, 0, 0, 0 };
        __builtin_amdgcn_tensor_load_to_lds(g0, g1, gz4, gz4, gz8, 0);
        __builtin_amdgcn_s_wait_tensorcnt(0);
    }
#endif

    // warm the near cache for the streaming operands (overlaps the tile DMA)
    {
        int pcol = bx * TX + (t & (TX - 1));
        int prow = by * TY + (t >> 6);
        int pi = prow * GRID_COLS + pcol;
        __builtin_prefetch(&w4[pi], 0, 3);
        __builtin_prefetch(&runoff[pi], 0, 3);
    }

#if USE_TDM
    __syncthreads();
#else
    for (int idx = t; idx < HY * HX; idx += 256) {
        int ly = idx / HX, lx = idx - ly * HX;
        int gr = startRow + ly, gc = startCol + lx;
        gr = gr < 0 ? 0 : (gr > GRID_ROWS - 1 ? GRID_ROWS - 1 : gr);
        gc = gc < 0 ? 0 : (gc > GRID_COLS - 1 ? GRID_COLS - 1 : gc);
        tile[idx] = qin[gr * GRID_COLS + gc];   // edge-clamped halo hits w==0
    }
    __syncthreads();
#endif

    const int tx = t & (TX - 1);
    const int ts = t >> 6;                 // 0..3, four rows per thread
    const int col = bx * TX + tx;
#pragma unroll
    for (int r = 0; r < 4; ++r) {
        int tyr = ts + 4 * r;              // 0..15
        int row = by * TY + tyr;
        int i  = row * GRID_COLS + col;
        int lc = (tyr + 1) * HX + (tx + 1);
        float4 w = w4[i];                  // one global_load_b128
        float inflow = w.x * tile[lc - 1]  + w.y * tile[lc + 1]
                     + w.z * tile[lc - HX] + w.w * tile[lc + HX];
        qout[i] = runoff[i] + inflow;
    }
}

// ---------------- kernel 1: phi, runoff, q0 init ----------------
__global__ __launch_bounds__(256) void ch_init(
    const float* __restrict__ melt, const float* __restrict__ bed,
    const float* __restrict__ wp,   const float* __restrict__ area,
    float* __restrict__ phi, float* __restrict__ runoff, float* __restrict__ q0)
{
    int i = blockIdx.x * 256 + threadIdx.x;
    float p = 9810.0f * bed[i] + wp[i];          // RHO_W * GRAV = 9810
    float r = (melt[i] * area[i]) / SEC_PER_A;
    phi[i] = p;
    runoff[i] = r;
    q0[i] = r;                                    // scan carry starts at runoff
}

// ---------------- kernel 2: gated reciprocal total drop ----------------
// g[i] = (core[i] && total_drop > 0) ? 1/total_drop : 0
__global__ __launch_bounds__(256) void ch_invdrop(
    const float* __restrict__ phi, const int* __restrict__ status,
    float* __restrict__ g)
{
    int i = blockIdx.x * 256 + threadIdx.x;
    int x = i & (GRID_COLS - 1);
    int y = i >> 10;
    float p = phi[i];
    float s = 0.0f;
    if (x > 0)             s += fmaxf(p - phi[i - 1], 0.0f);
    if (x < GRID_COLS - 1) s += fmaxf(p - phi[i + 1], 0.0f);
    if (y > 0)             s += fmaxf(p - phi[i - GRID_COLS], 0.0f);
    if (y < GRID_ROWS - 1) s += fmaxf(p - phi[i + GRID_COLS], 0.0f);
    g[i] = (status[i] == 0 && s > 0.0f) ? (1.0f / s) : 0.0f;
}

// ---------------- kernel 3: receiver-aligned inflow weights (packed) --------
// w4[i] = { from-left, from-right, from-up, from-down } inflow fractions.
__global__ __launch_bounds__(256) void ch_weights(
    const float* __restrict__ phi, const float* __restrict__ g,
    float4* __restrict__ w4)
{
    int i = blockIdx.x * 256 + threadIdx.x;
    int x = i & (GRID_COLS - 1);
    int y = i >> 10;
    float p = phi[i];
    float wl = (x > 0)             ? fmaxf(phi[i - 1] - p, 0.0f)         * g[i - 1]         : 0.0f;
    float wr = (x < GRID_COLS - 1) ? fmaxf(phi[i + 1] - p, 0.0f)         * g[i + 1]         : 0.0f;
    float wu = (y > 0)             ? fmaxf(phi[i - GRID_COLS] - p, 0.0f) * g[i - GRID_COLS] : 0.0f;
    float wd = (y < GRID_ROWS - 1) ? fmaxf(phi[i + GRID_COLS] - p, 0.0f) * g[i + GRID_COLS] : 0.0f;
    w4[i] = make_float4(wl, wr, wu, wd);
}

// ---------------- kernel 5: hydraulic gradient, masked to core ----------------
__global__ __launch_bounds__(256) void ch_final(
    const float* __restrict__ q, const float* __restrict__ conduit,
    const int* __restrict__ status, float* __restrict__ out)
{
    int i = blockIdx.x * 256 + threadIdx.x;
    float c = conduit[i];
    float c125 = c * __fsqrt_rn(__fsqrt_rn(c));      // c^(5/4), c > 0
    float v = q[i] * FLOW_COEFF * c125;
    out[i] = (status[i] == 0) ? v * v : 0.0f;
}

// ---------------- host launcher ----------------
extern "C" void kernel_launch(void* const* d_in, const int* in_sizes, int n_in,
                              void* d_out, int out_size, void* d_ws, size_t ws_size,
                              hipStream_t stream)
{
    const float* melt   = (const float*)d_in[0];
    const float* bed    = (const float*)d_in[1];
    const float* wp     = (const float*)d_in[2];
    const float* area   = (const float*)d_in[3];
    const float* cond   = (const float*)d_in[4];
    const int*   status = (const int*)  d_in[5];
    // d_in[6], d_in[7] (link_src/link_dst) encode the raster links; the grid
    // structure is derived analytically instead.

    float* ws     = (float*)d_ws;         // needs 9 * NNODES floats = 36 MB
    float*  phi    = ws;
    float*  runoff = ws + 1 * (size_t)NNODES;
    float*  g      = ws + 2 * (size_t)NNODES;
    float4* w4     = (float4*)(ws + 3 * (size_t)NNODES);   // 4*NNODES floats
    float*  q0     = ws + 7 * (size_t)NNODES;
    float*  q1     = ws + 8 * (size_t)NNODES;

    dim3 blk(256), grd(NNODES / 256);
    ch_init   <<<grd, blk, 0, stream>>>(melt, bed, wp, area, phi, runoff, q0);
    ch_invdrop<<<grd, blk, 0, stream>>>(phi, status, g);
    ch_weights<<<grd, blk, 0, stream>>>(phi, g, w4);

    dim3 sgrd(GRID_COLS / TX, GRID_ROWS / TY);
    float* qa = q0; float* qb = q1;
    for (int it = 0; it < N_ITERS; ++it) {
        ch_step<<<sgrd, blk, 0, stream>>>(qa, qb, runoff, w4);
        float* tmp = qa; qa = qb; qb = tmp;
    }
    ch_final<<<grd, blk, 0, stream>>>(qa, cond, status, (float*)d_out);
}